// CustomQuaternionLoss_1675037245666
// MI455X (gfx1250) — compile-verified
//
#include <hip/hip_runtime.h>
#include <math.h>

#define Q_DT   0.005f
#define Q_EPS  1e-9f

#define THREADS 256
#define NBLOCKS 512   // partial sums; B=524288 -> 4 rows/thread

// ---------------------------------------------------------------------------
// Pass 1: per-row quaternion loss, wave32 shuffle reduce -> LDS -> block partial
// ---------------------------------------------------------------------------
__global__ __launch_bounds__(THREADS)
void quat_loss_partial(const float* __restrict__ tq,    // [B,4] xyzw
                       const float* __restrict__ bias,  // [B,3]
                       const float* __restrict__ X,     // [B,S,6]
                       const float* __restrict__ qall,  // [N,4] wxyz
                       const float* __restrict__ Xmean, // [6]
                       const float* __restrict__ Xstd,  // [6]
                       const int*   __restrict__ idx,   // [B]
                       const int*   __restrict__ seqp,  // [1]
                       float*       __restrict__ partial,
                       int B)
{
    const int seq = seqp[0];
    // broadcast constants (scalar-cached)
    const float m3 = Xmean[3], m4 = Xmean[4], m5 = Xmean[5];
    const float s3 = Xstd[3],  s4 = Xstd[4],  s5 = Xstd[5];

    const int tid    = blockIdx.x * blockDim.x + threadIdx.x;
    const int stride = gridDim.x * blockDim.x;

    float acc = 0.0f;

    for (int b = tid; b < B; b += stride) {
        // ---- ang = batch_X[b, seq-1, 3:6] * std + mean  (streamed once -> NT)
        const float* xp = X + ((size_t)b * seq + (seq - 1)) * 6 + 3;
        const float ang0 = fmaf(__builtin_nontemporal_load(xp + 0), s3, m3);
        const float ang1 = fmaf(__builtin_nontemporal_load(xp + 1), s4, m4);
        const float ang2 = fmaf(__builtin_nontemporal_load(xp + 2), s5, m5);

        // ---- w_corr = ang - bias   (DISTURBANCE == 0)
        const float* bp = bias + (size_t)b * 3;
        const float w0 = ang0 - __builtin_nontemporal_load(bp + 0);
        const float w1 = ang1 - __builtin_nontemporal_load(bp + 1);
        const float w2 = ang2 - __builtin_nontemporal_load(bp + 2);

        // ---- correction quaternion  corr = [cos, sin * w/n]
        const float n  = sqrtf(w0 * w0 + w1 * w1 + w2 * w2);
        const float ht = n * (Q_DT * 0.5f);
        float sh, ch;
        sincosf(ht, &sh, &ch);
        const float sn = sh / n;                 // reference divides by n directly
        float cw = ch, cx = sn * w0, cy = sn * w1, cz = sn * w2;
        {   // normalize corr
            const float r = rsqrtf(cw * cw + cx * cx + cy * cy + cz * cz);
            cw *= r; cx *= r; cy *= r; cz *= r;
        }

        // ---- q0 = quaternions_all[max(idx - (seq-1), 0)]  (gather, keep cached)
        int ii = __builtin_nontemporal_load(idx + b) - (seq - 1);
        if (ii < 0) ii = 0;
        const float4 q0 = *(const float4*)(qall + (size_t)ii * 4);
        float qw = q0.x, qx = q0.y, qy = q0.z, qz = q0.w;
        {   // normalize q0
            const float r = rsqrtf(qw * qw + qx * qx + qy * qy + qz * qz);
            qw *= r; qx *= r; qy *= r; qz *= r;
        }

        // ---- q_comp = quat_mul_wxyz(q0n, corr)
        const float c0 = qw * cw - qx * cx - qy * cy - qz * cz;
        const float c1 = qw * cx + qx * cw + qy * cz - qz * cy;
        const float c2 = qw * cy - qx * cz + qy * cw + qz * cx;
        const float c3 = qw * cz + qx * cy - qy * cx + qz * cw;

        // ---- tq_conj (xyzw), streamed once -> NT float4
        const float4 t = {
            __builtin_nontemporal_load(tq + (size_t)b * 4 + 0),
            __builtin_nontemporal_load(tq + (size_t)b * 4 + 1),
            __builtin_nontemporal_load(tq + (size_t)b * 4 + 2),
            __builtin_nontemporal_load(tq + (size_t)b * 4 + 3)
        };
        const float x2 = -t.x, y2 = -t.y, z2 = -t.z, w2q = t.w;

        // ---- q_diff = quat_mul_xyzw(q_comp-as-xyzw, tq_conj)
        // reference reinterprets q_comp's components as x,y,z,w:
        const float x1 = c0, y1 = c1, z1 = c2, w1q = c3;
        float d0 = w1q * x2 + x1 * w2q + y1 * z2 - z1 * y2;
        float d1 = w1q * y2 - x1 * z2 + y1 * w2q + z1 * x2;
        float d2 = w1q * z2 + x1 * y2 - y1 * x2 + z1 * w2q;
        float d3 = w1q * w2q - x1 * x2 - y1 * y2 - z1 * z2;

        // ---- rotvec (log map)
        {
            const float r = rsqrtf(d0 * d0 + d1 * d1 + d2 * d2 + d3 * d3);
            d0 *= r; d1 *= r; d2 *= r; d3 *= r;
        }
        if (d3 < 0.0f) { d0 = -d0; d1 = -d1; d2 = -d2; d3 = -d3; }
        const float nv    = sqrtf(d0 * d0 + d1 * d1 + d2 * d2);
        const float angle = 2.0f * atan2f(nv, d3);
        const float scale = (nv < Q_EPS) ? 2.0f : angle / fmaxf(nv, Q_EPS);
        const float r0 = d0 * scale, r1 = d1 * scale, r2 = d2 * scale;

        // ---- Huber, accumulate
        const float a0 = fabsf(r0), a1 = fabsf(r1), a2 = fabsf(r2);
        acc += (a0 <= 1.0f) ? 0.5f * r0 * r0 : a0 - 0.5f;
        acc += (a1 <= 1.0f) ? 0.5f * r1 * r1 : a1 - 0.5f;
        acc += (a2 <= 1.0f) ? 0.5f * r2 * r2 : a2 - 0.5f;
    }

    // ---- wave32 reduce (gfx1250 is wave32-only)
    #pragma unroll
    for (int off = 16; off > 0; off >>= 1)
        acc += __shfl_down(acc, off, 32);

    __shared__ float lds[THREADS / 32];
    if ((threadIdx.x & 31) == 0) lds[threadIdx.x >> 5] = acc;
    __syncthreads();

    if (threadIdx.x < 32) {
        float v = (threadIdx.x < (THREADS / 32)) ? lds[threadIdx.x] : 0.0f;
        #pragma unroll
        for (int off = (THREADS / 64); off > 0; off >>= 1)
            v += __shfl_down(v, off, 32);
        if (threadIdx.x == 0) partial[blockIdx.x] = v;
    }
}

// ---------------------------------------------------------------------------
// Pass 2: deterministic tree reduction of the NBLOCKS partials, apply 1/(3B)
// ---------------------------------------------------------------------------
__global__ __launch_bounds__(NBLOCKS)
void quat_loss_final(const float* __restrict__ partial,
                     float* __restrict__ out,
                     float inv_count)
{
    __shared__ float lds[NBLOCKS];
    lds[threadIdx.x] = partial[threadIdx.x];
    __syncthreads();
    #pragma unroll
    for (int s = NBLOCKS / 2; s > 0; s >>= 1) {
        if (threadIdx.x < s) lds[threadIdx.x] += lds[threadIdx.x + s];
        __syncthreads();
    }
    if (threadIdx.x == 0) out[0] = lds[0] * inv_count;
}

// ---------------------------------------------------------------------------
extern "C" void kernel_launch(void* const* d_in, const int* in_sizes, int n_in,
                              void* d_out, int out_size, void* d_ws, size_t ws_size,
                              hipStream_t stream)
{
    // setup_inputs() order:
    // 0 true_quaternions [B,4] f32
    // 1 predicted_biases [B,3] f32
    // 2 batch_X          [B,S,6] f32
    // 3 quaternions_all  [N,4] f32
    // 4 X_mean           [1,1,6] f32
    // 5 X_std            [1,1,6] f32
    // 6 indices          [B] i32
    // 7 sequence_length  [1] i32
    const float* tq    = (const float*)d_in[0];
    const float* bias  = (const float*)d_in[1];
    const float* X     = (const float*)d_in[2];
    const float* qall  = (const float*)d_in[3];
    const float* Xmean = (const float*)d_in[4];
    const float* Xstd  = (const float*)d_in[5];
    const int*   idx   = (const int*)d_in[6];
    const int*   seqp  = (const int*)d_in[7];

    const int B = in_sizes[0] / 4;
    float* partial = (float*)d_ws;   // NBLOCKS floats of scratch

    quat_loss_partial<<<NBLOCKS, THREADS, 0, stream>>>(
        tq, bias, X, qall, Xmean, Xstd, idx, seqp, partial, B);

    const float inv_count = 1.0f / (3.0f * (float)B);
    quat_loss_final<<<1, NBLOCKS, 0, stream>>>(partial, (float*)d_out, inv_count);
}